// CoPE_11201274708510
// MI455X (gfx1250) — compile-verified
//
#include <hip/hip_runtime.h>

typedef __attribute__((ext_vector_type(2))) float v2f;
typedef __attribute__((ext_vector_type(4))) float v4f;
typedef __attribute__((ext_vector_type(8))) float v8f;

#define HH    16
#define NN    2048
#define DD    64
#define MAXP  64
#define NBITS 11   // log2(NN)

__global__ __launch_bounds__(256)
void cope_fused(const float* __restrict__ query,     // [H, N, D]
                const float* __restrict__ attn,      // [H, N, N]
                const float* __restrict__ pos_emb,   // [MAXP, D]
                float* __restrict__ out)             // [H, N, N]
{
    __shared__ float sTile[4][16][16];   // 4 p-tiles of the 16x16 WMMA C tile
    __shared__ float sLogits[MAXP];      // this row's 64 interaction logits
    __shared__ float sWaveSum[8];        // cross-wave scan combine

    const int row  = blockIdx.x;         // 0 .. H*N-1
    const int h    = row >> NBITS;
    const int n    = row & (NN - 1);
    const int t    = threadIdx.x;
    const int wid  = t >> 5;
    const int lane = t & 31;

    // ---------------- streaming load: 8 gates per thread (b128 x2, NT) ------
    const float* grow = attn + (size_t)row * NN;
    const int    base = t * 8;
    v4f g0 = __builtin_nontemporal_load((const v4f*)(grow + base));
    v4f g1 = __builtin_nontemporal_load((const v4f*)(grow + base + 4));

    // ---------------- WMMA: logits_int = q . pos_emb^T ----------------------
    // Waves 0..3: each computes one 16(M=n-tile) x 16(N=p-tile) f32 tile,
    // K = 64 via 16 steps of V_WMMA_F32_16X16X4_F32. Wave-uniform branch, so
    // EXEC is all-1s inside the WMMA chain as the ISA requires.
    if (wid < 4) {
        const int ptile = wid;
        const int ntile = n & ~15;
        const int m     = lane & 15;
        const int kh    = lane >> 4;     // 0: K={0,1}, 1: K={2,3} per A layout
        const float* qrow = query   + ((size_t)h * NN + (ntile + m)) * DD;
        const float* prow = pos_emb + (size_t)(ptile * 16 + m) * DD;

        // Prefetch all 16 A and 16 B fragments so the loads pipeline instead
        // of a load->wait(0)->wmma round trip per K-step.
        v2f af[16], bf[16];
#pragma unroll
        for (int k = 0; k < 16; ++k) {
            af[k] = *(const v2f*)(qrow + 4 * k + 2 * kh);
            bf[k] = *(const v2f*)(prow + 4 * k + 2 * kh);
        }

        v8f c = {};
#pragma unroll
        for (int k = 0; k < 16; ++k) {
            c = __builtin_amdgcn_wmma_f32_16x16x4_f32(
                    /*neg_a=*/false, af[k], /*neg_b=*/false, bf[k],
                    /*c_mod=*/(short)0, c, /*reuse_a=*/false, /*reuse_b=*/false);
        }

        // C layout: VGPR r -> M=r (lanes 0-15), M=r+8 (lanes 16-31); N = lane&15
        const int mBase = kh * 8;
        const int nc    = lane & 15;
#pragma unroll
        for (int r = 0; r < 8; ++r)
            sTile[ptile][mBase + r][nc] = c[r];
    }
    __syncthreads();

    // compact the one row we actually need into a flat 64-entry table
    if (t < MAXP)
        sLogits[t] = sTile[t >> 4][n & 15][t & 15];

    // ---------------- sigmoid + per-thread inclusive prefix (8 elems) -------
    auto sig = [](float x) {
        return __builtin_amdgcn_rcpf(1.0f + __expf(-x));   // v_exp + v_rcp
    };
    float s0 = sig(g0.x), s1 = sig(g0.y), s2 = sig(g0.z), s3 = sig(g0.w);
    float s4 = sig(g1.x), s5 = sig(g1.y), s6 = sig(g1.z), s7 = sig(g1.w);
    float i0 = s0;
    float i1 = i0 + s1;
    float i2 = i1 + s2;
    float i3 = i2 + s3;
    float i4 = i3 + s4;
    float i5 = i4 + s5;
    float i6 = i5 + s6;
    float i7 = i6 + s7;
    const float tot = i7;

    // ---------------- wave32 shuffle inclusive scan of thread totals --------
    float scan = tot;
#pragma unroll
    for (int d = 1; d < 32; d <<= 1) {
        float u = __shfl_up(scan, d, 32);
        if (lane >= d) scan += u;
    }
    if (lane == 31) sWaveSum[wid] = scan;
    __syncthreads();   // publishes sWaveSum AND sLogits

    float total = 0.0f, woff = 0.0f;
#pragma unroll
    for (int w = 0; w < 8; ++w) {
        float v = sWaveSum[w];
        total += v;
        woff  += (w < wid) ? v : 0.0f;
    }
    // exclusive prefix (elements strictly before this thread's first element)
    const float exclBase = woff + (scan - tot);

    // ---------------- interpolation: suffix = total - prefix_excl -----------
    const float* L = sLogits;
    auto interp = [&](float excl) {
        float pos = total - excl;
        pos = fminf(pos, 63.0f);
        pos = fmaxf(pos, 0.0f);
        float fl = floorf(pos);
        float w  = pos - fl;
        int ifl  = (int)fl;
        int icl  = ifl + (w > 0.0f ? 1 : 0);   // ceil without a second trans op
        float lf = L[ifl];
        float lc = L[icl];
        return lf + w * (lc - lf);
    };

    float r0 = interp(exclBase);
    float r1 = interp(exclBase + i0);
    float r2 = interp(exclBase + i1);
    float r3 = interp(exclBase + i2);
    float r4 = interp(exclBase + i3);
    float r5 = interp(exclBase + i4);
    float r6 = interp(exclBase + i5);
    float r7 = interp(exclBase + i6);

    float* orow = out + (size_t)row * NN;
    v4f o0 = {r0, r1, r2, r3};
    v4f o1 = {r4, r5, r6, r7};
    __builtin_nontemporal_store(o0, (v4f*)(orow + base));
    __builtin_nontemporal_store(o1, (v4f*)(orow + base + 4));
}

extern "C" void kernel_launch(void* const* d_in, const int* in_sizes, int n_in,
                              void* d_out, int out_size, void* d_ws, size_t ws_size,
                              hipStream_t stream) {
    (void)in_sizes; (void)n_in; (void)out_size; (void)d_ws; (void)ws_size;
    const float* query   = (const float*)d_in[0];   // [1,16,2048,64]
    const float* attn    = (const float*)d_in[1];   // [1,16,2048,2048]
    const float* pos_emb = (const float*)d_in[2];   // [64,64]
    float*       out     = (float*)d_out;           // [1,16,2048,2048]

    cope_fused<<<HH * NN, 256, 0, stream>>>(query, attn, pos_emb, out);
}